// SimCLRLoss_86174223827514
// MI455X (gfx1250) — compile-verified
//
#include <hip/hip_runtime.h>
#include <math.h>

typedef __attribute__((ext_vector_type(2))) float v2f;
typedef __attribute__((ext_vector_type(8))) float v8f;

#define N_HALF   4096
#define DIM      512
#define TWO_N    8192
#define INV_T    5.0f       // 1 / temperature
#define MAXLOGIT 5.0f       // |cos| <= 1  ->  logit <= 1/T = 5

// ---------------------------------------------------------------------------
// Kernel 1: inverse L2 norms of all 8192 rows (one wave32 per row).
// Also zero-initializes the scalar output accumulator.
// ---------------------------------------------------------------------------
__global__ __launch_bounds__(256) void simclr_norms_kernel(
    const float* __restrict__ Z1, const float* __restrict__ Z2,
    float* __restrict__ invn, float* __restrict__ out) {
  const int wave = threadIdx.x >> 5;
  const int lane = threadIdx.x & 31;
  const int row  = blockIdx.x * 8 + wave;   // 1024 blocks * 8 waves = 8192 rows

  const float* src = (row < N_HALF) ? (Z1 + (size_t)row * DIM)
                                    : (Z2 + (size_t)(row - N_HALF) * DIM);
  float ss = 0.0f;
  for (int c = lane; c < DIM; c += 32) {
    float v = src[c];
    ss += v * v;
  }
  ss += __shfl_xor(ss, 16, 32);
  ss += __shfl_xor(ss, 8, 32);
  ss += __shfl_xor(ss, 4, 32);
  ss += __shfl_xor(ss, 2, 32);
  ss += __shfl_xor(ss, 1, 32);
  if (lane == 0) invn[row] = 1.0f / fmaxf(sqrtf(ss), 1e-12f);

  if (blockIdx.x == 0 && threadIdx.x == 0) out[0] = 0.0f;
}

// ---------------------------------------------------------------------------
// Kernel 2: one block per 16-row strip of the 8192x8192 sim matrix.
//  - raw A strip (16 x 512 f32 = 32KB) staged in LDS
//  - 8 waves, each owning interleaved PAIRS of 16-col tiles (2 WMMA chains)
//  - fp32 WMMA 16x16x4, K swept in steps of 4 (128 steps)
//  - epilogue: x = dot * inv_i * inv_j * 5 ; acc += exp(x - 5);
//    diagonal / positive handling hoisted to (wave-uniform) tile level
// ---------------------------------------------------------------------------
__global__ __launch_bounds__(256) void simclr_main_kernel(
    const float* __restrict__ Z1, const float* __restrict__ Z2,
    const float* __restrict__ invn, float* __restrict__ out) {
  __shared__ float sA[16 * DIM];   // raw A strip (32KB)
  __shared__ float sInv[16];
  __shared__ float sSum[16];
  __shared__ float sPos[16];

  const int tid  = threadIdx.x;
  const int wave = tid >> 5;
  const int lane = tid & 31;
  const int half = lane >> 4;     // 0: lanes 0-15, 1: lanes 16-31
  const int l16  = lane & 15;
  const int r0   = blockIdx.x * 16;
  const int posb = (r0 + N_HALF) & (TWO_N - 1);  // 16-aligned tile of positives

  // --- stage raw A strip into LDS (strip never straddles the Z1/Z2 seam) ---
  const float* Asrc = (r0 < N_HALF) ? (Z1 + (size_t)r0 * DIM)
                                    : (Z2 + (size_t)(r0 - N_HALF) * DIM);
  {
    const float4* s4 = (const float4*)Asrc;
    float4* d4 = (float4*)sA;
    for (int i = tid; i < 16 * DIM / 4; i += 256) d4[i] = s4[i];
  }
  if (tid < 16) {
    sInv[tid] = invn[r0 + tid];
    sSum[tid] = 0.0f;
    sPos[tid] = 0.0f;
  }
  __syncthreads();

  float invr5[8];                 // row inv-norm with 1/T folded in
#pragma unroll
  for (int v = 0; v < 8; ++v) invr5[v] = sInv[v + half * 8] * INV_T;

  float acc[8];
#pragma unroll
  for (int v = 0; v < 8; ++v) acc[v] = 0.0f;

  // A fragment source: lane l<16 -> row l, K offsets {k,k+1};
  //                    lane l>=16 -> row l-16, K offsets {k+2,k+3}
  const float* sArow = sA + (size_t)l16 * DIM + half * 2;

  // 512 column tiles; each wave takes 2 adjacent tiles per pass, 8 waves
  for (int t = wave * 2; t < TWO_N / 16; t += 16) {
    const int c0 = t * 16;            // first tile's column base (c1 = c0+16)
    const float* B0 = ((c0 < N_HALF) ? (Z1 + (size_t)c0 * DIM)
                                     : (Z2 + (size_t)(c0 - N_HALF) * DIM))
                      + (size_t)l16 * DIM + half * 2;
    const int c1 = c0 + 16;
    const float* B1 = ((c1 < N_HALF) ? (Z1 + (size_t)c1 * DIM)
                                     : (Z2 + (size_t)(c1 - N_HALF) * DIM))
                      + (size_t)l16 * DIM + half * 2;

    const float invc0 = invn[c0 + l16];
    const float invc1 = invn[c1 + l16];

    v8f c0acc = {};
    v8f c1acc = {};
#pragma unroll 4
    for (int k = 0; k < DIM; k += 4) {
      v2f a, b0, b1;
      a[0]  = sArow[k]; a[1]  = sArow[k + 1];   // ds_load_2addr_b64
      b0[0] = B0[k];    b0[1] = B0[k + 1];      // global_load_b64 (L2 hit)
      b1[0] = B1[k];    b1[1] = B1[k + 1];
      c0acc = __builtin_amdgcn_wmma_f32_16x16x4_f32(
          false, a, false, b0, (short)0, c0acc, false, false);
      c1acc = __builtin_amdgcn_wmma_f32_16x16x4_f32(
          false, a, false, b1, (short)0, c1acc, false, false);
    }

    // C layout -> lane 0-15: (M=v, N=l16); lane 16-31: (M=v+8, N=l16)
    // Diagonal lives only in tile c==r0; positives only in tile c==posb.
    // Both conditions are wave-uniform -> tile-level scalar branch.
    const bool special = (c0 == r0) | (c0 == posb) | (c1 == r0) | (c1 == posb);

    if (!special) {                       // hot path: 508 of 512 tiles
#pragma unroll
      for (int v = 0; v < 8; ++v) {
        const float x0 = c0acc[v] * invr5[v] * invc0;
        const float x1 = c1acc[v] * invr5[v] * invc1;
        acc[v] += __expf(x0 - MAXLOGIT);
        acc[v] += __expf(x1 - MAXLOGIT);
      }
    } else {
#pragma unroll
      for (int v = 0; v < 8; ++v) {
        const int i = r0 + v + half * 8;
        const int jpos = (i + N_HALF) & (TWO_N - 1);
        {
          const int j = c0 + l16;
          const float x = c0acc[v] * invr5[v] * invc0;
          if (i != j) {
            if (j == jpos) sPos[v + half * 8] = x;   // unique writer
            acc[v] += __expf(x - MAXLOGIT);
          }
        }
        {
          const int j = c1 + l16;
          const float x = c1acc[v] * invr5[v] * invc1;
          if (i != j) {
            if (j == jpos) sPos[v + half * 8] = x;   // unique writer
            acc[v] += __expf(x - MAXLOGIT);
          }
        }
      }
    }
  }

  // reduce exp-sums across the 16 lanes of each half-wave, combine via LDS
#pragma unroll
  for (int v = 0; v < 8; ++v) {
    float a = acc[v];
    a += __shfl_xor(a, 1, 32);
    a += __shfl_xor(a, 2, 32);
    a += __shfl_xor(a, 4, 32);
    a += __shfl_xor(a, 8, 32);
    if (l16 == 0) atomicAdd(&sSum[v + half * 8], a);  // ds_add_f32
  }
  __syncthreads();

  if (tid < 16) {
    const float lse  = logf(sSum[tid]) + MAXLOGIT;
    const float loss = lse - sPos[tid];
    atomicAdd(out, loss * (1.0f / (float)TWO_N));     // global_atomic_add_f32
  }
}

// ---------------------------------------------------------------------------
extern "C" void kernel_launch(void* const* d_in, const int* in_sizes, int n_in,
                              void* d_out, int out_size, void* d_ws, size_t ws_size,
                              hipStream_t stream) {
  (void)in_sizes; (void)n_in; (void)out_size; (void)ws_size;
  const float* Z1 = (const float*)d_in[0];
  const float* Z2 = (const float*)d_in[1];
  float* invn = (float*)d_ws;       // 8192 floats = 32KB scratch
  float* out  = (float*)d_out;

  simclr_norms_kernel<<<TWO_N / 8, 256, 0, stream>>>(Z1, Z2, invn, out);
  simclr_main_kernel<<<TWO_N / 16, 256, 0, stream>>>(Z1, Z2, invn, out);
}